// LSGC_43413529428132
// MI455X (gfx1250) — compile-verified
//
#include <hip/hip_runtime.h>
#include <math.h>

typedef __attribute__((ext_vector_type(2))) float v2f;
typedef __attribute__((ext_vector_type(8))) float v8f;

#define EPS 1e-5f
constexpr int Bn = 32, Cc = 128, Hh = 64, Ww = 64;
constexpr int HW = Hh * Ww;                 // 4096
constexpr int CZ = 256;                     // channels of z = [x; fmax]
constexpr int NTRI = 136;                   // upper-triangular 16x16 tiles of S
constexpr float INV_N = 1.0f / (float)(Bn * HW);   // 1/131072

// ---- workspace float offsets ----
constexpr size_t F_OFF   = 0;                                  // fmax: 16,777,216 f
constexpr size_t S_OFF   = (size_t)Bn * Cc * HW;               // S (upper tri): 65536 f
constexpr size_t MUSUM_O = S_OFF + 65536;                      // 256
constexpr size_t MUZ_O   = MUSUM_O + 256;                      // 256
constexpr size_t ALPHA_O = MUZ_O + 256;                        // 256
constexpr size_t CVEC_O  = ALPHA_O + 256;                      // 256
constexpr size_t Q0_O    = CVEC_O + 256;                       // 32768
constexpr size_t Q0B_O   = Q0_O + 32768;                       // 128
constexpr size_t MUH_O   = Q0B_O + 128;                        // 128
constexpr size_t R_O     = MUH_O + 128;                        // 65536
constexpr size_t RB_O    = R_O + 65536;                        // 256
constexpr size_t COV_O   = RB_O + 256;                         // 65536 (full Cov)

__global__ void k_zero(float* __restrict__ p, int n) {
  int i = blockIdx.x * 256 + threadIdx.x;
  if (i < n) p[i] = 0.0f;
}

// fmax over circular-shift diffs; one (b,c) plane per block, staged in LDS.
__global__ void k_fmax(const float* __restrict__ x, float* __restrict__ f) {
  __shared__ float pl[HW];
  const int bc = blockIdx.x;
  const float* xp = x + (size_t)bc * HW;
  float* fp = f + (size_t)bc * HW;
  for (int i = threadIdx.x; i < HW; i += 256) pl[i] = xp[i];
  __syncthreads();
  for (int i = threadIdx.x; i < HW; i += 256) {
    int h = i >> 6, w = i & 63;
    float v = pl[i], m = 0.0f;  // step=64 gives zero diff -> max with 0
#pragma unroll
    for (int sh = 1; sh <= 5; ++sh) {
      int s = 1 << sh;  // 2,4,8,16,32
      m = fmaxf(m, v - pl[(h << 6) | ((w + s) & 63)]);
      m = fmaxf(m, v - pl[(h << 6) | ((w - s) & 63)]);
      m = fmaxf(m, v - pl[((((h + s) & 63)) << 6) | w]);
      m = fmaxf(m, v - pl[((((h - s) & 63)) << 6) | w]);
    }
    fp[i] = m;
  }
}

// per-channel sums of z = [x; f]
__global__ void k_chansum(const float* __restrict__ x, const float* __restrict__ f,
                          float* __restrict__ musum) {
  __shared__ float red[256];
  int cz = blockIdx.x, b = blockIdx.y;
  const float* p = (cz < Cc) ? (x + ((size_t)b * Cc + cz) * HW)
                             : (f + ((size_t)b * Cc + (cz - Cc)) * HW);
  float s = 0.0f;
  for (int i = threadIdx.x; i < HW; i += 256) s += p[i];
  red[threadIdx.x] = s; __syncthreads();
  for (int o = 128; o > 0; o >>= 1) {
    if (threadIdx.x < o) red[threadIdx.x] += red[threadIdx.x + o];
    __syncthreads();
  }
  if (threadIdx.x == 0) unsafeAtomicAdd(&musum[cz], red[0]);
}

// S = Z Z^T, upper-triangular 16x16 tiles only (symmetry), WMMA f32 16x16x4.
__global__ void k_syrk(const float* __restrict__ x, const float* __restrict__ f,
                       float* __restrict__ S) {
  // decode blockIdx.x in [0,136) -> (ti, tj) with ti <= tj
  int rem = blockIdx.x, ti = 0;
  while (rem >= 16 - ti) { rem -= 16 - ti; ++ti; }
  const int ib = ti << 4, jb = (ti + rem) << 4;
  const int b = blockIdx.y;
  const int lane = threadIdx.x & 31;
  const size_t pbase = (size_t)b * Cc * HW;
  const float* Az = (ib < Cc) ? x + pbase + (size_t)ib * HW : f + pbase + (size_t)(ib - Cc) * HW;
  const float* Bz = (jb < Cc) ? x + pbase + (size_t)jb * HW : f + pbase + (size_t)(jb - Cc) * HW;
  const int koff = (lane >> 4) << 1;  // 0 or 2
  const float* ap = Az + (size_t)(lane & 15) * HW + koff;
  const float* bp = Bz + (size_t)(lane & 15) * HW + koff;
  v8f acc = {};
  for (int k = 0; k < HW; k += 8) {
    v2f a0 = *(const v2f*)(ap + k);
    v2f b0 = *(const v2f*)(bp + k);
    v2f a1 = *(const v2f*)(ap + k + 4);
    v2f b1 = *(const v2f*)(bp + k + 4);
    acc = __builtin_amdgcn_wmma_f32_16x16x4_f32(false, a0, false, b0, (short)0, acc, false, false);
    acc = __builtin_amdgcn_wmma_f32_16x16x4_f32(false, a1, false, b1, (short)0, acc, false, false);
  }
  const int row0 = ib + ((lane >> 4) << 3);
  const int col = jb + (lane & 15);
#pragma unroll
  for (int r = 0; r < 8; ++r)
    unsafeAtomicAdd(&S[(size_t)(row0 + r) * CZ + col], acc[r]);
}

// Cov[c,d] = S[min,max]/N - mu_c mu_d  (full matrix, out-of-place); write mu_z.
__global__ void k_cov(const float* __restrict__ S, const float* __restrict__ musum,
                      float* __restrict__ Cov, float* __restrict__ muz) {
  int c = blockIdx.x, d = threadIdx.x;
  int lo = c < d ? c : d, hi = c < d ? d : c;
  float mc = musum[c] * INV_N, md = musum[d] * INV_N;
  Cov[(size_t)c * CZ + d] = S[(size_t)lo * CZ + hi] * INV_N - mc * md;
  if (d == 0) muz[c] = mc;
}

// Per-output BN1/BN2 fold: alpha[o], cvec[o] (= alpha*b + beta).
__global__ void k_solve1(const float* __restrict__ Cov, const float* __restrict__ muz,
                         const float* __restrict__ w_in, const float* __restrict__ b_in,
                         const float* __restrict__ g_ibn, const float* __restrict__ be_ibn,
                         const float* __restrict__ w_diff, const float* __restrict__ b_diff,
                         const float* __restrict__ g_bn, const float* __restrict__ be_bn,
                         float* __restrict__ alpha, float* __restrict__ cvec) {
  __shared__ float red[128], red2[128];
  int o = blockIdx.x, c = threadIdx.x;
  const float* Wr; float bias, g, be; int off;
  if (o < Cc) { Wr = w_in + (size_t)o * Cc; bias = b_in[o]; g = g_ibn[o]; be = be_ibn[o]; off = 0; }
  else { int oo = o - Cc; Wr = w_diff + (size_t)oo * Cc; bias = b_diff[oo]; g = g_bn[oo]; be = be_bn[oo]; off = Cc; }
  float wc = Wr[c];
  const float* crow = Cov + (size_t)(off + c) * CZ + off;
  float t = 0.0f;
  for (int d = 0; d < Cc; ++d) t += crow[d] * Wr[d];
  red[c] = wc * t;
  red2[c] = wc * muz[off + c];
  __syncthreads();
  for (int s = 64; s > 0; s >>= 1) {
    if (c < s) { red[c] += red[c + s]; red2[c] += red2[c + s]; }
    __syncthreads();
  }
  if (c == 0) {
    float var = red[0], muy = red2[0] + bias;
    float a = g * rsqrtf(var + EPS);
    float bta = be - a * muy;
    alpha[o] = a;
    cvec[o] = a * bias + bta;
  }
}

// Q0 = W_mr * diag(alpha) * blockdiag(W_in, W_diff); q0, mu_h.
__global__ void k_solve2(const float* __restrict__ alpha, const float* __restrict__ cvec,
                         const float* __restrict__ muz,
                         const float* __restrict__ w_in, const float* __restrict__ w_diff,
                         const float* __restrict__ w_mr, const float* __restrict__ b_mr,
                         float* __restrict__ Q0, float* __restrict__ q0, float* __restrict__ muh) {
  __shared__ float red[256];
  __shared__ float q0sh;
  int o = blockIdx.x, c = threadIdx.x;
  const float* Wm = w_mr + (size_t)o * CZ;
  float acc = 0.0f;
  if (c < Cc) {
    for (int k = 0; k < Cc; ++k) acc += Wm[k] * alpha[k] * w_in[(size_t)k * Cc + c];
  } else {
    int cc = c - Cc;
    for (int k = 0; k < Cc; ++k) acc += Wm[Cc + k] * alpha[Cc + k] * w_diff[(size_t)k * Cc + cc];
  }
  Q0[(size_t)o * CZ + c] = acc;
  red[c] = Wm[c] * cvec[c];
  __syncthreads();
  for (int s = 128; s > 0; s >>= 1) { if (c < s) red[c] += red[c + s]; __syncthreads(); }
  if (c == 0) { q0sh = red[0] + b_mr[o]; q0[o] = q0sh; }
  __syncthreads();
  red[c] = acc * muz[c];
  __syncthreads();
  for (int s = 128; s > 0; s >>= 1) { if (c < s) red[c] += red[c + s]; __syncthreads(); }
  if (c == 0) muh[o] = red[0] + q0sh;
}

// var_h = diag(Q0 Cov Q0^T); fold BN3; emit stacked R = [P; Q], rbias = [p; q].
__global__ void k_solve3(const float* __restrict__ Cov, const float* __restrict__ Q0,
                         const float* __restrict__ q0, const float* __restrict__ muh,
                         const float* __restrict__ alpha, const float* __restrict__ cvec,
                         const float* __restrict__ w_in,
                         const float* __restrict__ g_mbn, const float* __restrict__ be_mbn,
                         float* __restrict__ R, float* __restrict__ rbias) {
  __shared__ float red[256];
  __shared__ float a3sh;
  int o = blockIdx.x, c = threadIdx.x;
  const float* Qr = Q0 + (size_t)o * CZ;
  const float* crow = Cov + (size_t)c * CZ;
  float t = 0.0f;
  for (int d = 0; d < CZ; ++d) t += crow[d] * Qr[d];
  red[c] = Qr[c] * t;
  __syncthreads();
  for (int s = 128; s > 0; s >>= 1) { if (c < s) red[c] += red[c + s]; __syncthreads(); }
  if (c == 0) {
    float a3 = g_mbn[o] * rsqrtf(red[0] + EPS);
    float b3 = be_mbn[o] - a3 * muh[o];
    a3sh = a3;
    rbias[o] = cvec[o];                 // p
    rbias[Cc + o] = a3 * q0[o] + b3;    // q
  }
  __syncthreads();
  float a3 = a3sh;
  R[(size_t)o * CZ + c] = (c < Cc) ? alpha[o] * w_in[(size_t)o * Cc + c] : 0.0f;  // P row
  R[(size_t)(Cc + o) * CZ + c] = a3 * Qr[c];                                      // Q row
}

// out = P z + p + gelu(Q z + q). 8 waves/block; z tile staged once in LDS.
// P is zero on fmax channels, so cP only accumulates over k < 128.
__global__ void k_final(const float* __restrict__ x, const float* __restrict__ f,
                        const float* __restrict__ R, const float* __restrict__ rbias,
                        float* __restrict__ out) {
  __shared__ float zt[CZ * 16];  // [channel][col], 16 KB
  const int w = threadIdx.x >> 5, lane = threadIdx.x & 31;
  const int n0 = blockIdx.x << 4, b = blockIdx.y;
  const size_t pbase = (size_t)b * Cc * HW;

  // cooperative stage of z tile: channels 0..127 from x, 128..255 from f
  for (int i = threadIdx.x; i < Cc * 16; i += 256) {
    int c = i >> 4, col = i & 15;
    zt[i] = x[pbase + (size_t)c * HW + n0 + col];
    zt[Cc * 16 + i] = f[pbase + (size_t)c * HW + n0 + col];
  }
  __syncthreads();

  const int koff = (lane >> 4) << 1;
  const int col = lane & 15;
  const float* aP = R + (size_t)((w << 4) + col) * CZ + koff;
  const float* aQ = aP + (size_t)Cc * CZ;
  v8f cP = {}, cQ = {};
  for (int k = 0; k < Cc; k += 4) {      // x channels: both P and Q
    v2f a0 = *(const v2f*)(aP + k);
    v2f a1 = *(const v2f*)(aQ + k);
    int c0 = k + koff;
    v2f bb; bb.x = zt[(c0 << 4) | col]; bb.y = zt[((c0 + 1) << 4) | col];
    cP = __builtin_amdgcn_wmma_f32_16x16x4_f32(false, a0, false, bb, (short)0, cP, false, false);
    cQ = __builtin_amdgcn_wmma_f32_16x16x4_f32(false, a1, false, bb, (short)0, cQ, false, false);
  }
  for (int k = Cc; k < CZ; k += 4) {     // fmax channels: Q only (P half is zero)
    v2f a1 = *(const v2f*)(aQ + k);
    int c0 = k + koff;
    v2f bb; bb.x = zt[(c0 << 4) | col]; bb.y = zt[((c0 + 1) << 4) | col];
    cQ = __builtin_amdgcn_wmma_f32_16x16x4_f32(false, a1, false, bb, (short)0, cQ, false, false);
  }
  const int m0 = (lane >> 4) << 3;
#pragma unroll
  for (int r = 0; r < 8; ++r) {
    int orow = (w << 4) + m0 + r;
    float u = cP[r] + rbias[orow];
    float v = cQ[r] + rbias[Cc + orow];
    float gl = 0.5f * v * (1.0f + erff(v * 0.70710678118654752f));
    out[pbase + (size_t)orow * HW + n0 + col] = u + gl;
  }
}

extern "C" void kernel_launch(void* const* d_in, const int* in_sizes, int n_in,
                              void* d_out, int out_size, void* d_ws, size_t ws_size,
                              hipStream_t stream) {
  const float* x      = (const float*)d_in[0];
  const float* w_diff = (const float*)d_in[1];
  const float* b_diff = (const float*)d_in[2];
  const float* g_bn   = (const float*)d_in[3];
  const float* be_bn  = (const float*)d_in[4];
  const float* w_in   = (const float*)d_in[5];
  const float* b_in   = (const float*)d_in[6];
  const float* g_ibn  = (const float*)d_in[7];
  const float* be_ibn = (const float*)d_in[8];
  const float* w_mr   = (const float*)d_in[9];
  const float* b_mr   = (const float*)d_in[10];
  const float* g_mbn  = (const float*)d_in[11];
  const float* be_mbn = (const float*)d_in[12];

  float* ws    = (float*)d_ws;
  float* f     = ws + F_OFF;
  float* S     = ws + S_OFF;
  float* musum = ws + MUSUM_O;
  float* muz   = ws + MUZ_O;
  float* alpha = ws + ALPHA_O;
  float* cvec  = ws + CVEC_O;
  float* Q0    = ws + Q0_O;
  float* q0    = ws + Q0B_O;
  float* muh   = ws + MUH_O;
  float* R     = ws + R_O;
  float* rbias = ws + RB_O;
  float* Cov   = ws + COV_O;

  k_zero<<<257, 256, 0, stream>>>(S, 65536 + 256);            // S + musum (contiguous)
  k_fmax<<<Bn * Cc, 256, 0, stream>>>(x, f);
  k_chansum<<<dim3(CZ, Bn), 256, 0, stream>>>(x, f, musum);
  k_syrk<<<dim3(NTRI, Bn), 32, 0, stream>>>(x, f, S);
  k_cov<<<CZ, CZ, 0, stream>>>(S, musum, Cov, muz);
  k_solve1<<<CZ, 128, 0, stream>>>(Cov, muz, w_in, b_in, g_ibn, be_ibn,
                                   w_diff, b_diff, g_bn, be_bn, alpha, cvec);
  k_solve2<<<Cc, CZ, 0, stream>>>(alpha, cvec, muz, w_in, w_diff, w_mr, b_mr, Q0, q0, muh);
  k_solve3<<<Cc, CZ, 0, stream>>>(Cov, Q0, q0, muh, alpha, cvec, w_in, g_mbn, be_mbn, R, rbias);
  k_final<<<dim3(HW / 16, Bn), 256, 0, stream>>>(x, f, R, rbias, (float*)d_out);
}